// stock_lstm_85572928405562
// MI455X (gfx1250) — compile-verified
//
#include <hip/hip_runtime.h>
#include <stdint.h>

#define T_STEPS 524288
#define HID 8
#define CHUNK 2048                         // x elements per LDS buffer (8 KB)
#define NLOADS (CHUNK * 4 / (32 * 16))     // b128 per lane -> 512 B/instr -> 16

static_assert(T_STEPS % CHUNK == 0, "chunking");
static_assert(NLOADS == 16, "WAIT_ASYNC literal must match NLOADS");

#define L2E 1.4426950408889634f

#if __has_builtin(__builtin_amdgcn_exp2f)
#define EXP2F(x) __builtin_amdgcn_exp2f(x)
#else
#define EXP2F(x) exp2f(x)
#endif

#if __has_builtin(__builtin_amdgcn_rcpf)
#define RCPF(x) __builtin_amdgcn_rcpf(x)
#else
#define RCPF(x) (1.0f / (x))
#endif

// gfx1250 has v_tanh_f32. If the builtin exists, both tanh AND the sigmoids
// use it: sigmoid(z) = 0.5*tanh(z/2) + 0.5, with the /2 folded into the
// pre-scaled weights. Otherwise fall back to the exp2/rcp formulation with
// -L2E / -2*L2E pre-scales.
#if __has_builtin(__builtin_amdgcn_tanhf)
#define HAVE_TANH 1
#define TANHF(x) __builtin_amdgcn_tanhf(x)
#elif __has_builtin(__builtin_amdgcn_tanh_f32)
#define HAVE_TANH 1
#define TANHF(x) __builtin_amdgcn_tanh_f32(x)
#else
#define HAVE_TANH 0
#define TANHF(x) (fmaf(2.0f, RCPF(1.0f + EXP2F((x) * (-2.0f * L2E))), -1.0f))
#endif

// Constant-index lane broadcast: v_readlane_b32 -> SGPR (uniform), so the
// recurrent state h[0..7] lives in SGPRs and feeds v_fma as the scalar operand.
static __device__ __forceinline__ float lane_bcast(float v, int k) {
#if __has_builtin(__builtin_amdgcn_readlane)
  return __int_as_float(__builtin_amdgcn_readlane(__float_as_int(v), k));
#else
  return __int_as_float(__builtin_amdgcn_ds_bpermute(4 * k, __float_as_int(v)));
#endif
}

// CDNA5 async global->LDS copy (ASYNCcnt-tracked). One b128 per lane:
// 32 lanes * 16 B = 512 B per instruction; NLOADS instructions per chunk.
static __device__ __forceinline__ void async_load_chunk(const float* gsrc,
                                                        const float* lds_dst,
                                                        int lane) {
  unsigned ldsa = (unsigned)(uintptr_t)lds_dst + (unsigned)lane * 16u;
  unsigned long long ga = (unsigned long long)(uintptr_t)gsrc +
                          (unsigned long long)lane * 16ull;
#pragma unroll
  for (int i = 0; i < NLOADS; ++i) {
    asm volatile("global_load_async_to_lds_b128 %0, %1, off"
                 :: "v"(ldsa + (unsigned)(i * 512)),
                    "v"(ga + (unsigned long long)(i * 512))
                 : "memory");
  }
}

#define WAIT_ASYNC(n) asm volatile("s_wait_asynccnt " #n ::: "memory")

__global__ __launch_bounds__(32, 1)
void lstm_serial_kernel(const float* __restrict__ x,
                        const float* __restrict__ h0,
                        const float* __restrict__ c0,
                        const float* __restrict__ W_ih,
                        const float* __restrict__ W_hh,
                        const float* __restrict__ b_ih,
                        const float* __restrict__ b_hh,
                        const float* __restrict__ W_out,
                        float* __restrict__ out) {
  __shared__ float xs[2][CHUNK];
  const int lane = threadIdx.x;   // one wave32
  const int kj = lane & 7;        // hidden unit owned by this lane (4x replicated)

  // Every lane computes ALL FOUR gates (i,f,g,o = rows kj+0/8/16/24) of its
  // hidden unit -> the whole gate->cell->hidden update is lane-local; the only
  // cross-lane traffic is the 8-value h broadcast (v_readlane, VALU).
  float wih[4], bias[4], whh[4][HID];
#pragma unroll
  for (int q = 0; q < 4; ++q) {
    const int row = kj + 8 * q;
#if HAVE_TANH
    const float s = (q == 2) ? 1.0f : 0.5f;            // tanh(z) / 0.5*tanh(z/2)+0.5
#else
    const float s = (q == 2) ? (-2.0f * L2E) : (-L2E); // exp2-based fallback
#endif
    wih[q]  = W_ih[row] * s;                           // W_ih is [32,1]
    bias[q] = (b_ih[row] + b_hh[row]) * s;
#pragma unroll
    for (int k = 0; k < HID; ++k) whh[q][k] = W_hh[row * HID + k] * s;
  }
  float wout[HID];
#pragma unroll
  for (int k = 0; k < HID; ++k) wout[k] = W_out[k];

  // State: h broadcast (uniform, SGPR-resident after readlane); c per kj.
  float h[HID];
#pragma unroll
  for (int k = 0; k < HID; ++k) h[k] = h0[k];
  float c = c0[kj];

  async_load_chunk(x, xs[0], lane);   // prime buffer 0

  float ybuf = 0.0f;

  for (int base = 0; base < T_STEPS; base += CHUNK) {
    const int buf = (base / CHUNK) & 1;
    if (base + CHUNK < T_STEPS) {
      async_load_chunk(x + base + CHUNK, xs[buf ^ 1], lane);
      WAIT_ASYNC(16);   // <= NLOADS outstanding => current chunk complete
    } else {
      WAIT_ASYNC(0);
    }

    for (int tb = 0; tb < CHUNK; tb += 32) {
      float4 xq;
#pragma unroll
      for (int u = 0; u < 32; ++u) {
        if ((u & 3) == 0)  // broadcast ds_load_b128: 4 steps of x at once
          xq = *reinterpret_cast<const float4*>(&xs[buf][tb + u]);
        const float xt = ((u & 3) == 0) ? xq.x
                       : ((u & 3) == 1) ? xq.y
                       : ((u & 3) == 2) ? xq.z : xq.w;

        // Four pre-scaled gate pre-activations; x/bias term is h-independent
        // so the compiler hoists it off the recurrent critical path.
        float g[4];
#pragma unroll
        for (int q = 0; q < 4; ++q) {
          float ga = fmaf(xt, wih[q], bias[q]);
          ga = fmaf(whh[q][0], h[0], ga);
          ga = fmaf(whh[q][2], h[2], ga);
          ga = fmaf(whh[q][4], h[4], ga);
          ga = fmaf(whh[q][6], h[6], ga);
          float gb = whh[q][1] * h[1];
          gb = fmaf(whh[q][3], h[3], gb);
          gb = fmaf(whh[q][5], h[5], gb);
          gb = fmaf(whh[q][7], h[7], gb);
          g[q] = ga + gb;
        }

#if HAVE_TANH
        // One v_tanh_f32 per gate; sigmoid = 0.5*tanh(z/2)+0.5 (scale folded).
        const float ri = fmaf(0.5f, TANHF(g[0]), 0.5f);  // sigmoid(i)
        const float rf = fmaf(0.5f, TANHF(g[1]), 0.5f);  // sigmoid(f)
        const float gg = TANHF(g[2]);                    // tanh(g)
        const float ro = fmaf(0.5f, TANHF(g[3]), 0.5f);  // sigmoid(o)
#else
        const float ri = RCPF(1.0f + EXP2F(g[0]));
        const float rf = RCPF(1.0f + EXP2F(g[1]));
        const float gg = fmaf(2.0f, RCPF(1.0f + EXP2F(g[2])), -1.0f);
        const float ro = RCPF(1.0f + EXP2F(g[3]));
#endif

        // Cell/hidden update — fully lane-local.
        c = fmaf(rf, c, ri * gg);
        const float hm = ro * TANHF(c);

        // Broadcast h[0..7] (lanes 0..7 hold them) via v_readlane -> SGPRs.
#pragma unroll
        for (int k = 0; k < HID; ++k) h[k] = lane_bcast(hm, k);

        // Output projection (uniform, off the serial critical path).
        const float ya = fmaf(h[0], wout[0], h[1] * wout[1]);
        const float yb = fmaf(h[2], wout[2], h[3] * wout[3]);
        const float yc = fmaf(h[4], wout[4], h[5] * wout[5]);
        const float yd = fmaf(h[6], wout[6], h[7] * wout[7]);
        const float y  = (ya + yb) + (yc + yd);
        if (lane == u) ybuf = y;   // bank result for coalesced store
      }
      out[base + tb + lane] = ybuf;  // 32 results, one global_store_b32
    }
  }
}

extern "C" void kernel_launch(void* const* d_in, const int* in_sizes, int n_in,
                              void* d_out, int out_size, void* d_ws, size_t ws_size,
                              hipStream_t stream) {
  (void)in_sizes; (void)n_in; (void)out_size; (void)d_ws; (void)ws_size;
  const float* x     = (const float*)d_in[0];
  const float* h0    = (const float*)d_in[1];
  const float* c0    = (const float*)d_in[2];
  const float* W_ih  = (const float*)d_in[3];
  const float* W_hh  = (const float*)d_in[4];
  const float* b_ih  = (const float*)d_in[5];
  const float* b_hh  = (const float*)d_in[6];
  const float* W_out = (const float*)d_in[7];
  float* out = (float*)d_out;

  // One wave32: the LSTM recurrence is strictly serial; all state in VGPRs/SGPRs.
  lstm_serial_kernel<<<dim3(1), dim3(32), 0, stream>>>(
      x, h0, c0, W_ih, W_hh, b_ih, b_hh, W_out, out);
}